// OrderGPT2_SentencePresetS2IP_76304388981035
// MI455X (gfx1250) — compile-verified
//
#include <hip/hip_runtime.h>
#include <hip/hip_bf16.h>
#include <math.h>

// Problem constants (match reference)
#define BB 512
#define LL 128
#define HH 768
#define AA 65536
#define VV 32000
#define KK 32

typedef __attribute__((ext_vector_type(2))) float v2f;
typedef __attribute__((ext_vector_type(8))) float v8f;

// ---------------------------------------------------------------------------
// Fused gather + mean over L, plus last-token embedding.
//   query[b,h]    = mean_l order_emb[x[b,l], h]
//   emb_last[b,h] = order_emb[x[b,L-1], h]
// ---------------------------------------------------------------------------
__global__ __launch_bounds__(256) void gather_mean_last_kernel(
    const int* __restrict__ x, const float* __restrict__ emb_table,
    float* __restrict__ query, float* __restrict__ emb_last) {
  __shared__ int xs[LL];
  const int b = blockIdx.x;
  const int t = threadIdx.x;
  if (t < LL) xs[t] = x[b * LL + t];
  __syncthreads();

  const int h0 = t, h1 = t + 256, h2 = t + 512;  // 768 = 3 * 256
  float a0 = 0.f, a1 = 0.f, a2 = 0.f;
  for (int l = 0; l < LL; ++l) {
    const float* r = emb_table + (size_t)xs[l] * HH;
    a0 += r[h0];
    a1 += r[h1];
    a2 += r[h2];
  }
  const float inv = 1.0f / (float)LL;
  float* qo = query + (size_t)b * HH;
  qo[h0] = a0 * inv;
  qo[h1] = a1 * inv;
  qo[h2] = a2 * inv;

  const float* rl = emb_table + (size_t)xs[LL - 1] * HH;
  float* eo = emb_last + (size_t)b * HH;
  eo[h0] = rl[h0];
  eo[h1] = rl[h1];
  eo[h2] = rl[h2];
}

// ---------------------------------------------------------------------------
// WMMA fp32 GEMM:  C[M,N] = (X[M,K] @ W[N,K]^T) * row_scale[m] * col_scale[n] + bias[n]
// row_scale / col_scale / bias may be nullptr.
// Block = 256 threads = 8 waves; wave tile 64x32 (8 x v8f accumulators),
// block tile 256x64. Uses V_WMMA_F32_16X16X4_F32.
//
// Per-lane fragment addressing for 16x16x4 f32 (ISA 7.12.2):
//   A(m,k): lane = 16*(k>=2) + m, vgpr = k&1  -> lane reads X[m][kb], X[m][kb+1]
//   B(k,n): lane = 16*(k>=2) + n, vgpr = k&1  -> lane reads W[n][kb], W[n][kb+1]
//   C(m,n): vgpr i, lanes 0-15: m=i, lanes 16-31: m=i+8; n = lane&15
// ---------------------------------------------------------------------------
__global__ __launch_bounds__(256) void gemm_wmma_f32_kernel(
    const float* __restrict__ X, const float* __restrict__ W,
    float* __restrict__ C, int M, int N, int Kd,
    const float* __restrict__ row_scale, const float* __restrict__ col_scale,
    const float* __restrict__ bias) {
  const int lane = threadIdx.x & 31;
  const int wave = threadIdx.x >> 5;
  const int wm = wave & 3;   // 4 wave-rows
  const int wn = wave >> 2;  // 2 wave-cols
  const int row0 = blockIdx.y * 256 + wm * 64;
  const int col0 = blockIdx.x * 64 + wn * 32;

  const int ml = lane & 15;         // m (A) or n (B) within 16-subtile
  const int kb = (lane >> 4) * 2;   // k-pair selected by upper lane half

  const float* ap[4];
  const float* bp[2];
#pragma unroll
  for (int mi = 0; mi < 4; ++mi)
    ap[mi] = X + (size_t)(row0 + 16 * mi + ml) * Kd + kb;
#pragma unroll
  for (int ni = 0; ni < 2; ++ni)
    bp[ni] = W + (size_t)(col0 + 16 * ni + ml) * Kd + kb;

  v8f acc[4][2];
#pragma unroll
  for (int mi = 0; mi < 4; ++mi)
#pragma unroll
    for (int ni = 0; ni < 2; ++ni)
      acc[mi][ni] = (v8f){0.f, 0.f, 0.f, 0.f, 0.f, 0.f, 0.f, 0.f};

#pragma unroll 2
  for (int k = 0; k < Kd; k += 4) {
    v2f af[4], bf[2];
#pragma unroll
    for (int mi = 0; mi < 4; ++mi) af[mi] = *(const v2f*)(ap[mi] + k);
#pragma unroll
    for (int ni = 0; ni < 2; ++ni) bf[ni] = *(const v2f*)(bp[ni] + k);
#pragma unroll
    for (int mi = 0; mi < 4; ++mi)
#pragma unroll
      for (int ni = 0; ni < 2; ++ni)
        acc[mi][ni] = __builtin_amdgcn_wmma_f32_16x16x4_f32(
            /*neg_a=*/false, af[mi], /*neg_b=*/false, bf[ni],
            /*c_mod=*/(short)0, acc[mi][ni], /*reuse_a=*/false,
            /*reuse_b=*/false);
  }

  const int mo = (lane >> 4) * 8;  // m offset within a 16-row subtile
  const int nl = lane & 15;
#pragma unroll
  for (int mi = 0; mi < 4; ++mi) {
#pragma unroll
    for (int ni = 0; ni < 2; ++ni) {
      const int ncol = col0 + 16 * ni + nl;
      const float cs = col_scale ? col_scale[ncol] : 1.0f;
      const float bv = bias ? bias[ncol] : 0.0f;
#pragma unroll
      for (int i = 0; i < 8; ++i) {
        const int mrow = row0 + 16 * mi + mo + i;
        const float rs = row_scale ? row_scale[mrow] : 1.0f;
        C[(size_t)mrow * N + ncol] = acc[mi][ni][i] * rs * cs + bv;
      }
    }
  }
}

// ---------------------------------------------------------------------------
// Reciprocal row L2-norms with torch-F.normalize semantics:
//   out[r] = 1 / max(||X[r,:]||_2, 1e-12)
// One wave per row; 8 rows per block.
// ---------------------------------------------------------------------------
__global__ __launch_bounds__(256) void row_rnorm_kernel(
    const float* __restrict__ X, float* __restrict__ out, int rows) {
  const int lane = threadIdx.x & 31;
  const int wave = threadIdx.x >> 5;
  const int row = blockIdx.x * 8 + wave;
  if (row >= rows) return;
  const float* r = X + (size_t)row * HH;
  float s = 0.f;
  for (int j = lane; j < HH; j += 32) {
    const float v = r[j];
    s += v * v;
  }
#pragma unroll
  for (int off = 16; off > 0; off >>= 1) s += __shfl_xor(s, off, 32);
  if (lane == 0) out[row] = 1.0f / fmaxf(sqrtf(s), 1e-12f);
}

// ---------------------------------------------------------------------------
// Per-row top-K (K=32) of A=65536 scores. One block of 256 threads per row.
// Phase 1: each thread keeps the top-32 of its strided 256-element slice
//          (the union of per-thread top-32s contains the row's global top-32).
// Phase 2: 32 block-wide argmax extractions over the 8192 LDS candidates,
//          emitting values in descending order (matches jax.lax.top_k).
// Also accumulates sum(topk) for align_mean.
// ---------------------------------------------------------------------------
#define NCAND (256 * KK)
__global__ __launch_bounds__(256) void topk_row_kernel(
    const float* __restrict__ scores, float* __restrict__ out_idx,
    float* __restrict__ out_val, float* __restrict__ align_sum) {
  __shared__ float lv[NCAND];
  __shared__ int li[NCAND];
  __shared__ float rv[256];
  __shared__ int rp[256];

  const int row = blockIdx.x;
  const int t = threadIdx.x;
  const float* srow = scores + (size_t)row * AA;
  const float NEG = -__builtin_inff();

  float v[KK];
  int id[KK];
#pragma unroll
  for (int j = 0; j < KK; ++j) {
    v[j] = NEG;
    id[j] = 0x7fffffff;
  }
  float vmin = NEG;
  int pm = 0;
  for (int i = 0; i < AA / 256; ++i) {
    const int idx = i * 256 + t;  // coalesced across threads
    const float e = srow[idx];
    if (e > vmin) {
      v[pm] = e;
      id[pm] = idx;
      vmin = v[0];
      pm = 0;
#pragma unroll
      for (int j = 1; j < KK; ++j)
        if (v[j] < vmin) {
          vmin = v[j];
          pm = j;
        }
    }
  }
#pragma unroll
  for (int j = 0; j < KK; ++j) {
    lv[t * KK + j] = v[j];
    li[t * KK + j] = id[j];
  }
  __syncthreads();

  float mysum = 0.f;
  for (int it = 0; it < KK; ++it) {
    // local argmax over this thread's 32 candidate slots
    float bm = NEG;
    int bp = -1;
#pragma unroll
    for (int j = 0; j < KK; ++j) {
      const int slot = t * KK + j;
      const float e = lv[slot];
      if (e > bm || (e == bm && bp >= 0 && li[slot] < li[bp])) {
        bm = e;
        bp = slot;
      }
    }
    rv[t] = bm;
    rp[t] = bp;
    __syncthreads();
    // block argmax (ties -> smaller original index)
    for (int s = 128; s > 0; s >>= 1) {
      if (t < s) {
        const float e = rv[t + s];
        const int p2 = rp[t + s];
        const bool take =
            (e > rv[t]) ||
            (e == rv[t] && p2 >= 0 && (rp[t] < 0 || li[p2] < li[rp[t]]));
        if (take) {
          rv[t] = e;
          rp[t] = p2;
        }
      }
      __syncthreads();
    }
    if (t == 0) {
      const int p = rp[0];
      out_val[row * KK + it] = rv[0];
      out_idx[row * KK + it] = (float)li[p];
      mysum += rv[0];
      lv[p] = NEG;  // invalidate winner
    }
    __syncthreads();
  }
  if (t == 0) atomicAdd(align_sum, mysum);
}

__global__ void zero_scalar_kernel(float* p) { p[0] = 0.0f; }

__global__ void finalize_mean_kernel(const float* __restrict__ s,
                                     float* __restrict__ out) {
  out[0] = s[0] * (1.0f / (float)(BB * KK));
}

// ---------------------------------------------------------------------------
// kernel_launch
// Outputs (flat, in reference return order, all fp32):
//   logits [B*V] | scores [B*A] | align_mean [1] | topk_idx [B*K] | topk_scores [B*K]
// ---------------------------------------------------------------------------
extern "C" void kernel_launch(void* const* d_in, const int* in_sizes, int n_in,
                              void* d_out, int out_size, void* d_ws,
                              size_t ws_size, hipStream_t stream) {
  const int* x = (const int*)d_in[0];
  const float* order_emb = (const float*)d_in[1];
  const float* anchor_raw = (const float*)d_in[2];
  const float* Wq = (const float*)d_in[3];
  const float* Wa = (const float*)d_in[4];
  const float* W_trunk = (const float*)d_in[5];
  const float* head_W = (const float*)d_in[6];
  const float* head_b = (const float*)d_in[7];
  float* out = (float*)d_out;

  const size_t off_logits = 0;
  const size_t off_scores = (size_t)BB * VV;
  const size_t off_align = off_scores + (size_t)BB * AA;
  const size_t off_tidx = off_align + 1;
  const size_t off_tval = off_tidx + (size_t)BB * KK;

  // Workspace layout (floats)
  float* ws = (float*)d_ws;
  float* query = ws;                               // B*H
  float* emb_last = query + (size_t)BB * HH;       // B*H
  float* qbuf = emb_last + (size_t)BB * HH;        // B*H
  float* hbuf = qbuf + (size_t)BB * HH;            // B*H
  float* rq = hbuf + (size_t)BB * HH;              // B
  float* ra = rq + BB;                             // A
  float* asum = ra + AA;                           // 1
  float* abuf = asum + 1;                          // A*H (192 MB, biggest last)

  const dim3 blk(256);

  zero_scalar_kernel<<<1, 1, 0, stream>>>(asum);

  // emb gather + mean + last token
  gather_mean_last_kernel<<<BB, blk, 0, stream>>>(x, order_emb, query,
                                                  emb_last);

  // q = query @ Wq^T                 [512 x 768]
  gemm_wmma_f32_kernel<<<dim3(HH / 64, BB / 256), blk, 0, stream>>>(
      query, Wq, qbuf, BB, HH, HH, nullptr, nullptr, nullptr);

  // a = anchor_raw @ Wa^T            [65536 x 768]  (dominant GEMM)
  gemm_wmma_f32_kernel<<<dim3(HH / 64, AA / 256), blk, 0, stream>>>(
      anchor_raw, Wa, abuf, AA, HH, HH, nullptr, nullptr, nullptr);

  // reciprocal L2 norms for fused normalization
  row_rnorm_kernel<<<BB / 8, blk, 0, stream>>>(qbuf, rq, BB);
  row_rnorm_kernel<<<AA / 8, blk, 0, stream>>>(abuf, ra, AA);

  // scores = (q @ a^T) * rq[m] * ra[n]   [512 x 65536], written straight to d_out
  gemm_wmma_f32_kernel<<<dim3(AA / 64, BB / 256), blk, 0, stream>>>(
      qbuf, abuf, out + off_scores, BB, AA, HH, rq, ra, nullptr);

  // top-32 per row + align_mean accumulation (scores still L2-resident)
  topk_row_kernel<<<BB, blk, 0, stream>>>(out + off_scores, out + off_tidx,
                                          out + off_tval, asum);

  // Only the last token of gpt_input feeds the logits:
  // h_last = emb_last @ W_trunk^T    [512 x 768]
  gemm_wmma_f32_kernel<<<dim3(HH / 64, BB / 256), blk, 0, stream>>>(
      emb_last, W_trunk, hbuf, BB, HH, HH, nullptr, nullptr, nullptr);

  // logits = h_last @ head_W^T + head_b   [512 x 32000]
  gemm_wmma_f32_kernel<<<dim3(VV / 64, BB / 256), blk, 0, stream>>>(
      hbuf, head_W, out + off_logits, BB, VV, HH, nullptr, nullptr, head_b);

  finalize_mean_kernel<<<1, 1, 0, stream>>>(asum, out + off_align);
}